// ProteinMPNN_15899968930126
// MI455X (gfx1250) — compile-verified
//
#include <hip/hip_runtime.h>
#include <math.h>

// ---------------- problem constants ----------------
#define NB      2
#define NN      1024
#define KNB     48       // neighbors per node (= 3 M-tiles of 16)
#define HH      128
#define HID     512
#define K3H     384      // 3*H edge-MLP input
#define SCALE_INV (1.0f/30.0f)
#define LN_EPS  1e-5f

// ---------------- LDS layout (floats) ----------------
#define XS      388                  // stride of X rows (even, padded)
#define HS      132                  // stride of activation rows (even, padded)
#define XOFF    0                    // X: 48 x 384  (also reused for H2)
#define HOFF    (KNB*XS)             // H1: 48 x 128     = 18624
#define MROW    (HOFF + KNB*HS)      // mask_attend row  = 24960
#define RED     (MROW + KNB)         // 2 x 128 partials = 25008
#define NODE_HV (RED + 256)          // original hV copy = 25264
#define NODE_XN (NODE_HV + HH)       // working node vec = 25392
#define NODE_HD (NODE_XN + HH)       // FFN hidden 512   = 25520
#define NODE_ST (NODE_HD + HID)      // mu, invstd       = 26032
#define LDS_FLOATS 26048             // ~104 KB

typedef float v2f __attribute__((ext_vector_type(2)));
typedef float v8f __attribute__((ext_vector_type(8)));

__device__ __forceinline__ float gelu_f(float x) {
    return 0.5f * x * (1.0f + erff(x * 0.70710678118654752f));
}

// D(16x16) += A(16x4) * B(4x16), fp32 WMMA, 3 stacked M-tiles (48 rows).
// A-frag layout: lanes 0-15 -> M=lane, K=kb..kb+1 ; lanes 16-31 -> M=lane-16, K=kb+2..kb+3
// B-frag layout: v0/v1 = rows K(+0/+1) for lanes 0-15, K(+2/+3) for lanes 16-31; col = n0+(lane&15)
__device__ __forceinline__ void gemm3(const float* __restrict__ src, int stride, int Kdim,
                                      const float* __restrict__ W, int n0, int lane, v8f acc[3]) {
    const int half2 = (lane >> 4) << 1;     // 0 or 2
    const int col   = n0 + (lane & 15);
    const int rowb  = lane & 15;
    for (int kb = 0; kb < Kdim; kb += 4) {
        const int k2 = kb + half2;
        v2f bw;
        bw.x = W[(size_t)k2 * HH + col];
        bw.y = W[(size_t)(k2 + 1) * HH + col];
        const float* s0 = src + k2;
#pragma unroll
        for (int m = 0; m < 3; ++m) {
            v2f aw = *(const v2f*)(s0 + (m * 16 + rowb) * stride);
            acc[m] = __builtin_amdgcn_wmma_f32_16x16x4_f32(
                false, aw, false, bw, (short)0, acc[m], false, false);
        }
    }
}

// C/D layout: VGPR v -> row = mtile*16 + v + 8*(lane>=16); col = n0+(lane&15)
__device__ __forceinline__ void store_act(float* __restrict__ dst, int stride, const v8f acc[3],
                                          const float* __restrict__ bias, int n0, int lane,
                                          bool do_gelu) {
    const int col   = n0 + (lane & 15);
    const int rbase = (lane >> 4) * 8;
    const float bc  = bias[col];
#pragma unroll
    for (int m = 0; m < 3; ++m)
#pragma unroll
        for (int v = 0; v < 8; ++v) {
            float x = acc[m][v] + bc;
            if (do_gelu) x = gelu_f(x);
            dst[(m * 16 + rbase + v) * stride + col] = x;
        }
}

__device__ __forceinline__ void zero3(v8f acc[3]) {
#pragma unroll
    for (int m = 0; m < 3; ++m) acc[m] = (v8f){0, 0, 0, 0, 0, 0, 0, 0};
}

// Stage h_EV = [hV_i | hE_ij | hV_j] into LDS X (48 x 384)
__device__ __forceinline__ void load_X(float* __restrict__ lds, const float* __restrict__ hV,
                                       const float* __restrict__ hE, const int* __restrict__ E_idx,
                                       int bn, int tid) {
    const size_t eb    = (size_t)bn * KNB;
    const int    bbase = (bn >> 10) << 10;       // start of this batch's nodes
    const float* hVi   = hV + (size_t)bn * HH;
    for (int i = tid; i < KNB * HH; i += 256) {
        const int r = i >> 7, c = i & 127;
        lds[XOFF + r * XS + c] = hVi[c];
        lds[XOFF + r * XS + HH + c] = hE[(eb + r) * HH + c];
        const int j = E_idx[eb + r];
        lds[XOFF + r * XS + 2 * HH + c] = hV[((size_t)(bbase + j)) * HH + c];
    }
}

struct NodeP {
    const float *W1, *b1, *W2, *b2, *W3, *b3;
    const float *Wi, *bi, *Wo, *bo, *n1s, *n1b, *n2s, *n2b;
};
struct EdgeP {
    const float *W1, *b1, *W2, *b2, *W3, *b3, *n3s, *n3b;
};

// ---------- Kernel A: edge-message MLP + neighbor sum + node update ----------
__global__ __launch_bounds__(256) void node_layer_kernel(
    const float* __restrict__ hV_in, float* __restrict__ hV_out,
    const float* __restrict__ hE, const int* __restrict__ E_idx,
    const float* __restrict__ mask, const float* __restrict__ mask_attend, NodeP p) {
    __shared__ __align__(16) float lds[LDS_FLOATS];
    const int tid = threadIdx.x, lane = tid & 31, wave = tid >> 5;
    const int n0  = wave << 4;
    const int bn  = blockIdx.x;

    load_X(lds, hV_in, hE, E_idx, bn, tid);
    if (tid < HH)  lds[NODE_HV + tid] = hV_in[(size_t)bn * HH + tid];
    if (tid < KNB) lds[MROW + tid] = mask_attend[(size_t)bn * KNB + tid];
    __syncthreads();

    v8f acc[3];
    zero3(acc);
    gemm3(lds + XOFF, XS, K3H, p.W1, n0, lane, acc);         // 384 -> 128
    store_act(lds + HOFF, HS, acc, p.b1, n0, lane, true);
    __syncthreads();
    zero3(acc);
    gemm3(lds + HOFF, HS, HH, p.W2, n0, lane, acc);          // 128 -> 128
    store_act(lds + XOFF, HS, acc, p.b2, n0, lane, true);    // H2 reuses X region
    __syncthreads();
    zero3(acc);
    gemm3(lds + XOFF, HS, HH, p.W3, n0, lane, acc);          // 128 -> 128

    // masked partial column sums over this lane's 24 rows
    {
        const int col = n0 + (lane & 15);
        const int rbase = (lane >> 4) * 8;
        const float bc = p.b3[col];
        float part = 0.f;
#pragma unroll
        for (int m = 0; m < 3; ++m)
#pragma unroll
            for (int v = 0; v < 8; ++v) {
                const int row = m * 16 + rbase + v;
                part += (acc[m][v] + bc) * lds[MROW + row];
            }
        lds[RED + (lane >> 4) * HH + col] = part;
    }
    __syncthreads();
    if (tid < HH) {
        const float dh = (lds[RED + tid] + lds[RED + HH + tid]) * SCALE_INV;
        lds[NODE_XN + tid] = lds[NODE_HV + tid] + dh;
    }
    __syncthreads();
    if (wave == 0) {                                          // LN1 stats
        float s = 0.f, s2 = 0.f;
#pragma unroll
        for (int q = 0; q < 4; ++q) { float v = lds[NODE_XN + lane + 32 * q]; s += v; s2 += v * v; }
        for (int off = 16; off; off >>= 1) { s += __shfl_xor(s, off, 32); s2 += __shfl_xor(s2, off, 32); }
        if (lane == 0) {
            const float mu = s * (1.f / 128.f);
            lds[NODE_ST] = mu;
            lds[NODE_ST + 1] = rsqrtf(s2 * (1.f / 128.f) - mu * mu + LN_EPS);
        }
    }
    __syncthreads();
    if (tid < HH) {
        const float mu = lds[NODE_ST], inv = lds[NODE_ST + 1];
        lds[NODE_XN + tid] = (lds[NODE_XN + tid] - mu) * inv * p.n1s[tid] + p.n1b[tid];
    }
    __syncthreads();
    for (int j = tid; j < HID; j += 256) {                    // FFN 128 -> 512
        float s = p.bi[j];
        for (int c = 0; c < HH; ++c) s += lds[NODE_XN + c] * p.Wi[(size_t)c * HID + j];
        lds[NODE_HD + j] = gelu_f(s);
    }
    __syncthreads();
    if (tid < HH) {                                           // FFN 512 -> 128 + residual
        float s = p.bo[tid];
        for (int j = 0; j < HID; ++j) s += lds[NODE_HD + j] * p.Wo[(size_t)j * HH + tid];
        lds[RED + tid] = lds[NODE_XN + tid] + s;
    }
    __syncthreads();
    if (wave == 0) {                                          // LN2 stats
        float s = 0.f, s2 = 0.f;
#pragma unroll
        for (int q = 0; q < 4; ++q) { float v = lds[RED + lane + 32 * q]; s += v; s2 += v * v; }
        for (int off = 16; off; off >>= 1) { s += __shfl_xor(s, off, 32); s2 += __shfl_xor(s2, off, 32); }
        if (lane == 0) {
            const float mu = s * (1.f / 128.f);
            lds[NODE_ST] = mu;
            lds[NODE_ST + 1] = rsqrtf(s2 * (1.f / 128.f) - mu * mu + LN_EPS);
        }
    }
    __syncthreads();
    if (tid < HH) {
        const float mu = lds[NODE_ST], inv = lds[NODE_ST + 1];
        const float y = (lds[RED + tid] - mu) * inv * p.n2s[tid] + p.n2b[tid];
        hV_out[(size_t)bn * HH + tid] = mask[bn] * y;
    }
}

// ---------- Kernel B (encoder only): edge update, hE = LN(hE + MLP(h_EV)) ----------
__global__ __launch_bounds__(256) void edge_layer_kernel(
    const float* __restrict__ hV, float* __restrict__ hE,
    const int* __restrict__ E_idx, EdgeP p) {
    __shared__ __align__(16) float lds[LDS_FLOATS];
    const int tid = threadIdx.x, lane = tid & 31, wave = tid >> 5;
    const int n0  = wave << 4;
    const int bn  = blockIdx.x;

    load_X(lds, hV, hE, E_idx, bn, tid);
    __syncthreads();

    v8f acc[3];
    zero3(acc);
    gemm3(lds + XOFF, XS, K3H, p.W1, n0, lane, acc);
    store_act(lds + HOFF, HS, acc, p.b1, n0, lane, true);
    __syncthreads();
    zero3(acc);
    gemm3(lds + HOFF, HS, HH, p.W2, n0, lane, acc);
    store_act(lds + XOFF, HS, acc, p.b2, n0, lane, true);
    __syncthreads();
    zero3(acc);
    gemm3(lds + XOFF, HS, HH, p.W3, n0, lane, acc);
    store_act(lds + HOFF, HS, acc, p.b3, n0, lane, false);    // m -> H1 region
    __syncthreads();

    // per-edge (row) LayerNorm of hE + m, wave-parallel over columns
    const size_t eb = (size_t)bn * KNB;
    for (int r = wave; r < KNB; r += 8) {
        float* hErow = hE + (eb + r) * HH;
        float vals[4];
        float s = 0.f, s2 = 0.f;
#pragma unroll
        for (int q = 0; q < 4; ++q) {
            const int c = lane + 32 * q;
            const float v = hErow[c] + lds[HOFF + r * HS + c];
            vals[q] = v; s += v; s2 += v * v;
        }
        for (int off = 16; off; off >>= 1) { s += __shfl_xor(s, off, 32); s2 += __shfl_xor(s2, off, 32); }
        const float mu  = s * (1.f / 128.f);
        const float inv = rsqrtf(s2 * (1.f / 128.f) - mu * mu + LN_EPS);
#pragma unroll
        for (int q = 0; q < 4; ++q) {
            const int c = lane + 32 * q;
            hErow[c] = (vals[q] - mu) * inv * p.n3s[c] + p.n3b[c];
        }
    }
}

// ---------------- host side ----------------
// Input order (JAX pytree = sorted dict keys inside each params dict):
// 0 h_V, 1 h_E, 2 E_idx, 3 mask, 4 mask_attend,
// enc (base 5):  W1 W11 W12 W13 W2 W3 Wi Wo b1 b11 b12 b13 b2 b3 bi bo n1b n1s n2b n2s n3b n3s
// dec (base 27): W1 W2 W3 Wi Wo b1 b2 b3 bi bo n1b n1s n2b n2s
#define EI(k) ((const float*)d_in[5 + (k)])
#define DI(k) ((const float*)d_in[27 + (k)])

extern "C" void kernel_launch(void* const* d_in, const int* in_sizes, int n_in,
                              void* d_out, int out_size, void* d_ws, size_t ws_size,
                              hipStream_t stream) {
    (void)in_sizes; (void)n_in; (void)out_size; (void)ws_size;

    const int nBlocks = NB * NN;                 // 2048 (one block per node)
    const size_t hvN = (size_t)NB * NN * HH;     // 262144 floats
    const size_t heN = (size_t)NB * NN * KNB * HH;

    float* hVa = (float*)d_ws;
    float* hVb = hVa + hvN;
    float* hEw = hVb + hvN;

    // working copy of hE (encoder updates it in place)
    hipMemcpyAsync(hEw, d_in[1], heN * sizeof(float), hipMemcpyDeviceToDevice, stream);

    const int*   E_idx = (const int*)d_in[2];
    const float* mask  = (const float*)d_in[3];
    const float* matt  = (const float*)d_in[4];

    const size_t W1L = (size_t)K3H * HH, WsL = (size_t)HH * HH;
    const size_t WiL = (size_t)HH * HID, WoL = (size_t)HID * HH;

    const float* hVcur = (const float*)d_in[0];
    float* bufs[2] = {hVa, hVb};
    int tog = 0;

    for (int l = 0; l < 3; ++l) {                // encoder
        NodeP np;
        np.W1 = EI(0) + l * W1L;  np.b1 = EI(8)  + l * HH;
        np.W2 = EI(4) + l * WsL;  np.b2 = EI(12) + l * HH;
        np.W3 = EI(5) + l * WsL;  np.b3 = EI(13) + l * HH;
        np.Wi = EI(6) + l * WiL;  np.bi = EI(14) + l * HID;
        np.Wo = EI(7) + l * WoL;  np.bo = EI(15) + l * HH;
        np.n1s = EI(17) + l * HH; np.n1b = EI(16) + l * HH;
        np.n2s = EI(19) + l * HH; np.n2b = EI(18) + l * HH;

        float* hVnext = bufs[tog]; tog ^= 1;
        node_layer_kernel<<<nBlocks, 256, 0, stream>>>(hVcur, hVnext, hEw, E_idx, mask, matt, np);

        EdgeP ep;
        ep.W1 = EI(1) + l * W1L;  ep.b1 = EI(9)  + l * HH;
        ep.W2 = EI(2) + l * WsL;  ep.b2 = EI(10) + l * HH;
        ep.W3 = EI(3) + l * WsL;  ep.b3 = EI(11) + l * HH;
        ep.n3s = EI(21) + l * HH; ep.n3b = EI(20) + l * HH;
        edge_layer_kernel<<<nBlocks, 256, 0, stream>>>(hVnext, hEw, E_idx, ep);

        hVcur = hVnext;
    }

    for (int l = 0; l < 3; ++l) {                // decoder
        NodeP np;
        np.W1 = DI(0) + l * W1L;  np.b1 = DI(5) + l * HH;
        np.W2 = DI(1) + l * WsL;  np.b2 = DI(6) + l * HH;
        np.W3 = DI(2) + l * WsL;  np.b3 = DI(7) + l * HH;
        np.Wi = DI(3) + l * WiL;  np.bi = DI(8) + l * HID;
        np.Wo = DI(4) + l * WoL;  np.bo = DI(9) + l * HH;
        np.n1s = DI(11) + l * HH; np.n1b = DI(10) + l * HH;
        np.n2s = DI(13) + l * HH; np.n2b = DI(12) + l * HH;

        float* hVnext = (l == 2) ? (float*)d_out : bufs[tog];
        tog ^= 1;
        node_layer_kernel<<<nBlocks, 256, 0, stream>>>(hVcur, hVnext, hEw, E_idx, mask, matt, np);
        hVcur = hVnext;
    }
}